// DiscriminativeLoss_54820962566853
// MI455X (gfx1250) — compile-verified
//
#include <hip/hip_runtime.h>

typedef float v2f __attribute__((ext_vector_type(2)));
typedef float v8f __attribute__((ext_vector_type(8)));

#define BATCH     16
#define EMBED_DIM 16
#define NPIX      (256 * 512)
#define NLANES    6
#define DELTA_V   0.5f
#define DELTA_D   3.0f

// Work partitioning
#define CHUNKS        64                 // x-blocks per batch
#define PIX_PER_BLK   (NPIX / CHUNKS)    // 2048
#define PIX_PER_WAVE  (PIX_PER_BLK / 8)  // 256 (8 waves / 256-thread block)
#define WMMA_ITERS    (PIX_PER_WAVE / 4) // 64 (K=4 pixels per WMMA)
#define VAR_ITERS     (PIX_PER_BLK / 256)// 8

// ws layout (floats)
// [0,1536)      sums[b][l][d]
// [1536,1632)   counts[b][l]
// [1632,3168)   means[b][l][d]
// [3168,3264)   varsum[b][l]
#define WS_SUMS   0
#define WS_COUNTS 1536
#define WS_MEANS  1632
#define WS_VARSUM 3168
#define WS_TOTAL  3264

__global__ void zero_ws_kernel(float* __restrict__ p, int n) {
  int i = blockIdx.x * blockDim.x + threadIdx.x;
  if (i < n) p[i] = 0.0f;
}

// ---- pass 0: per-(b,l) pixel counts -------------------------------------
__global__ void __launch_bounds__(256)
count_kernel(const int* __restrict__ seg, float* __restrict__ counts) {
  const int b = blockIdx.y;
  const int base = blockIdx.x * PIX_PER_BLK;
  __shared__ float c[NLANES];
  if (threadIdx.x < NLANES) c[threadIdx.x] = 0.0f;
  __syncthreads();

  float local[NLANES] = {0.f, 0.f, 0.f, 0.f, 0.f, 0.f};
  const int* sp = seg + (size_t)b * NPIX + base;
#pragma unroll
  for (int it = 0; it < VAR_ITERS; ++it) {
    int s = sp[it * 256 + threadIdx.x];
#pragma unroll
    for (int l = 0; l < NLANES; ++l) local[l] += (s == l + 1) ? 1.0f : 0.0f;
  }
#pragma unroll
  for (int l = 0; l < NLANES; ++l) atomicAdd(&c[l], local[l]);
  __syncthreads();
  if (threadIdx.x < NLANES) atomicAdd(&counts[b * NLANES + threadIdx.x], c[threadIdx.x]);
}

// ---- pass 1: centroid sums via f32 WMMA (D=16 x L onehot contraction) ---
// C[M=d][N=l] += sum_k emb[b][d][p+k] * (seg[b][p+k] == l+1)
__global__ void __launch_bounds__(256)
means_wmma_kernel(const float* __restrict__ emb, const int* __restrict__ seg,
                  float* __restrict__ sums) {
  const int b     = blockIdx.y;
  const int wave  = threadIdx.x >> 5;
  const int lane  = threadIdx.x & 31;
  const int col   = lane & 15;   // A: row M=d    B: column N=l
  const int khalf = lane >> 4;   // K-half select per ISA 32-bit layouts

  const int p0 = blockIdx.x * PIX_PER_BLK + wave * PIX_PER_WAVE;

  const float* aptr = emb + ((size_t)(b * EMBED_DIM + col)) * NPIX + p0 + khalf * 2;
  const int*   sptr = seg + (size_t)b * NPIX + p0 + khalf * 2;

  v8f c = {0.f, 0.f, 0.f, 0.f, 0.f, 0.f, 0.f, 0.f};

  for (int it = 0; it < WMMA_ITERS; ++it) {
    // A 16x4 f32: vgpr0 = K=0 (lanes 0-15) / K=2 (lanes 16-31); vgpr1 = K=1 / K=3
    v2f a = *(const v2f*)aptr;
    int s0 = sptr[0];
    int s1 = sptr[1];
    // B 4x16 f32 one-hot, same K-half striping
    v2f bb;
    bb.x = (s0 == col + 1) ? 1.0f : 0.0f;
    bb.y = (s1 == col + 1) ? 1.0f : 0.0f;
    c = __builtin_amdgcn_wmma_f32_16x16x4_f32(false, a, false, bb, (short)0, c,
                                              false, false);
    aptr += 4;
    sptr += 4;
  }

  // block-level reduction of the 8 wave accumulators
  __shared__ float tile[16][16];
  ((float*)tile)[threadIdx.x] = 0.0f;
  __syncthreads();
#pragma unroll
  for (int r = 0; r < 8; ++r) {
    int d = r + khalf * 8;  // C layout: lanes 0-15 -> M=r, lanes 16-31 -> M=8+r
    atomicAdd(&tile[d][col], c[r]);
  }
  __syncthreads();
  if (threadIdx.x < NLANES * 16) {
    int l = threadIdx.x >> 4;
    int d = threadIdx.x & 15;
    atomicAdd(&sums[(b * NLANES + l) * 16 + d], tile[d][l]);
  }
}

// ---- pass 2: means = sums / max(count,1) --------------------------------
__global__ void finalize_means_kernel(const float* __restrict__ sums,
                                      const float* __restrict__ counts,
                                      float* __restrict__ means) {
  int idx = blockIdx.x * 256 + threadIdx.x;
  if (idx >= BATCH * NLANES * 16) return;
  float cnt = counts[idx >> 4];
  means[idx] = sums[idx] / fmaxf(cnt, 1.0f);
}

// ---- pass 3: variance (pull) hinge, binned per (b,l) --------------------
__global__ void __launch_bounds__(256)
var_kernel(const float* __restrict__ emb, const int* __restrict__ seg,
           const float* __restrict__ means, float* __restrict__ varsum) {
  const int b = blockIdx.y;
  const int base = blockIdx.x * PIX_PER_BLK;
  __shared__ float ls[NLANES];
  if (threadIdx.x < NLANES) ls[threadIdx.x] = 0.0f;
  __syncthreads();

  float acc[NLANES] = {0.f, 0.f, 0.f, 0.f, 0.f, 0.f};
  const int*   sp = seg + (size_t)b * NPIX + base;
  const float* ep = emb + (size_t)b * EMBED_DIM * NPIX + base;
  const float* mp = means + b * NLANES * 16;

#pragma unroll
  for (int it = 0; it < VAR_ITERS; ++it) {
    int n = it * 256 + threadIdx.x;
    int s = sp[n];
    if (s > 0) {
      const float* m = mp + (s - 1) * 16;
      float sq = 0.0f;
#pragma unroll
      for (int d = 0; d < EMBED_DIM; ++d) {
        float t = ep[(size_t)d * NPIX + n] - m[d];
        sq += t * t;
      }
      float norm = sqrtf(sq);
      float h = fmaxf(norm - DELTA_V, 0.0f);
      float hv = h * h;
#pragma unroll
      for (int l = 0; l < NLANES; ++l) acc[l] += (s == l + 1) ? hv : 0.0f;
    }
  }
#pragma unroll
  for (int l = 0; l < NLANES; ++l) atomicAdd(&ls[l], acc[l]);
  __syncthreads();
  if (threadIdx.x < NLANES) atomicAdd(&varsum[b * NLANES + threadIdx.x], ls[threadIdx.x]);
}

// ---- pass 4: per-batch combine + pairwise push loss + mean over B -------
__global__ void final_kernel(const float* __restrict__ counts,
                             const float* __restrict__ varsum,
                             const float* __restrict__ means,
                             float* __restrict__ out) {
  __shared__ float red[32];
  const int t = threadIdx.x;
  float contrib = 0.0f;
  if (t < BATCH) {
    const float* cnt = counts + t * NLANES;
    const float* vs  = varsum + t * NLANES;
    const float* mn  = means + t * NLANES * 16;
    float num = 0.0f, vsum = 0.0f;
    for (int l = 0; l < NLANES; ++l) {
      float c = cnt[l];
      if (c > 0.0f) {
        num += 1.0f;
        vsum += vs[l] / c;
      }
    }
    float var_b = (num > 0.0f) ? vsum / num : 0.0f;

    float psum = 0.0f;
    for (int i = 0; i < NLANES; ++i) {
      for (int j = i + 1; j < NLANES; ++j) {
        if (cnt[i] > 0.0f && cnt[j] > 0.0f) {
          float d2 = 0.0f;
          for (int d = 0; d < 16; ++d) {
            float df = mn[i * 16 + d] - mn[j * 16 + d];
            d2 += df * df;
          }
          float dist = sqrtf(d2);
          float h = fmaxf(DELTA_D - dist, 0.0f);
          psum += h * h;
        }
      }
    }
    // reference: full-offdiag-sum / (num*(num-1)) / 2  ==  i<j sum / (num*(num-1))
    float dist_b = (num > 1.0f) ? psum / (num * (num - 1.0f)) : 0.0f;
    contrib = var_b + dist_b;
  }
  red[t] = contrib;
  __syncthreads();
  if (t == 0) {
    float s = 0.0f;
    for (int i = 0; i < 32; ++i) s += red[i];
    out[0] = s / (float)BATCH;
  }
}

extern "C" void kernel_launch(void* const* d_in, const int* in_sizes, int n_in,
                              void* d_out, int out_size, void* d_ws, size_t ws_size,
                              hipStream_t stream) {
  const float* emb = (const float*)d_in[0];
  const int*   seg = (const int*)d_in[1];
  float* ws = (float*)d_ws;
  float* out = (float*)d_out;

  float* sums   = ws + WS_SUMS;
  float* counts = ws + WS_COUNTS;
  float* means  = ws + WS_MEANS;
  float* varsum = ws + WS_VARSUM;

  zero_ws_kernel<<<(WS_TOTAL + 255) / 256, 256, 0, stream>>>(ws, WS_TOTAL);
  count_kernel<<<dim3(CHUNKS, BATCH), 256, 0, stream>>>(seg, counts);
  means_wmma_kernel<<<dim3(CHUNKS, BATCH), 256, 0, stream>>>(emb, seg, sums);
  finalize_means_kernel<<<(BATCH * NLANES * 16 + 255) / 256, 256, 0, stream>>>(
      sums, counts, means);
  var_kernel<<<dim3(CHUNKS, BATCH), 256, 0, stream>>>(emb, seg, means, varsum);
  final_kernel<<<1, 32, 0, stream>>>(counts, varsum, means, out);
}